// RiemannianStyleLoss_75015898792419
// MI455X (gfx1250) — compile-verified
//
#include <hip/hip_runtime.h>

typedef __attribute__((ext_vector_type(2))) float v2f;
typedef __attribute__((ext_vector_type(8))) float v8f;

#define NPIX    16384
#define SCALEF  0.13025f
// base-4 packed (dy+1), (dx+1) for the 9 shifts (0,0)(0,1)(0,-1)(1,0)(-1,0)(1,1)(1,-1)(-1,1)(-1,-1)
#define DY_PACK 10389
#define DX_PACK 35145

// ---------------------------------------------------------------------------
// Kernel 1: per (input,batch) 48x48-padded Gram via V_WMMA_F32_16X16X4_F32,
// then mean-corrected covariance written to ws (36x36 per matrix).
// ---------------------------------------------------------------------------
__global__ __launch_bounds__(256) void gram_cov_kernel(const float* __restrict__ x,
                                                       const float* __restrict__ y,
                                                       float* __restrict__ covOut) {
    __shared__ float G[48 * 48];
    __shared__ float csum[4];
    const int tid = threadIdx.x;
    const int b   = blockIdx.x & 63;
    const int inp = blockIdx.x >> 6;
    const float* __restrict__ src = (inp ? y : x) + (size_t)b * 4 * NPIX;

    for (int i = tid; i < 48 * 48; i += 256) G[i] = 0.0f;
    if (tid < 4) csum[tid] = 0.0f;
    __syncthreads();

    // channel sums (rolls are permutations -> shifted means == base means)
    #pragma unroll
    for (int c = 0; c < 4; ++c) {
        const float* __restrict__ p = src + (c << 14);
        float partial = 0.0f;
        for (int e = tid; e < NPIX; e += 256) partial += p[e];
        atomicAdd(&csum[c], partial);
    }

    const int lane  = tid & 31;
    const int wv    = tid >> 5;           // 8 waves, each owns interleaved K-chunks
    const int mrow  = lane & 15;          // M/N index within a 16-wide tile
    const int phase = (lane >> 4) << 1;   // lanes 16-31 carry K+2,K+3

    // per-lane, per-tile channel decode: ch = t*16 + mrow ; ch = s*4 + c
    int dy[3], dx[3], cb[3], valid[3];
    #pragma unroll
    for (int t = 0; t < 3; ++t) {
        int ch   = t * 16 + mrow;
        valid[t] = (ch < 36);
        int s    = ch >> 2;
        dy[t] = valid[t] ? (((DY_PACK >> (2 * s)) & 3) - 1) : 0;
        dx[t] = valid[t] ? (((DX_PACK >> (2 * s)) & 3) - 1) : 0;
        cb[t] = (ch & 3) << 14;
    }

    v8f a00 = {}, a01 = {}, a02 = {}, a11 = {}, a12 = {}, a22 = {};

    for (int it = 0; it < NPIX / 32; ++it) {
        const int p0 = it * 32 + wv * 4 + phase;   // p0, p0+1 never cross a row
        const int i0 = p0 >> 7;
        const int j0 = p0 & 127;
        v2f av[3];
        #pragma unroll
        for (int t = 0; t < 3; ++t) {
            int ii  = (i0 - dy[t]) & 127;
            int jj0 = (j0 - dx[t]) & 127;
            int jj1 = (j0 + 1 - dx[t]) & 127;
            const float* __restrict__ cp = src + cb[t] + (ii << 7);
            av[t].x = valid[t] ? (SCALEF * cp[jj0]) : 0.0f;
            av[t].y = valid[t] ? (SCALEF * cp[jj1]) : 0.0f;
        }
        // Gram: A-format tile == B-format tile, so reuse registers for A and B.
        a00 = __builtin_amdgcn_wmma_f32_16x16x4_f32(false, av[0], false, av[0], (short)0, a00, false, false);
        a01 = __builtin_amdgcn_wmma_f32_16x16x4_f32(false, av[0], false, av[1], (short)0, a01, false, false);
        a02 = __builtin_amdgcn_wmma_f32_16x16x4_f32(false, av[0], false, av[2], (short)0, a02, false, false);
        a11 = __builtin_amdgcn_wmma_f32_16x16x4_f32(false, av[1], false, av[1], (short)0, a11, false, false);
        a12 = __builtin_amdgcn_wmma_f32_16x16x4_f32(false, av[1], false, av[2], (short)0, a12, false, false);
        a22 = __builtin_amdgcn_wmma_f32_16x16x4_f32(false, av[2], false, av[2], (short)0, a22, false, false);
    }

    // reduce wave-partial Grams into LDS (C/D layout: VGPR r -> M=r / r+8)
    const int rowoff = (lane >> 4) << 3;
    const int col    = lane & 15;
    {
        const v8f* accs[6] = { &a00, &a01, &a02, &a11, &a12, &a22 };
        const int  tmv[6]  = { 0, 0, 0, 1, 1, 2 };
        const int  tnv[6]  = { 0, 1, 2, 1, 2, 2 };
        #pragma unroll
        for (int pIdx = 0; pIdx < 6; ++pIdx) {
            #pragma unroll
            for (int r = 0; r < 8; ++r) {
                int row = tmv[pIdx] * 16 + r + rowoff;
                atomicAdd(&G[row * 48 + tnv[pIdx] * 16 + col], (*accs[pIdx])[r]);
            }
        }
    }
    __syncthreads();

    // cov[i][j] = (G - N*mi*mj)/(N-1) + 1e-5*I  (only upper-tri tile pairs stored)
    const float invN = 1.0f / (float)NPIX;
    const float norm = 1.0f / (float)(NPIX - 1);
    float* __restrict__ out = covOut + (size_t)blockIdx.x * 1296;
    for (int idx = tid; idx < 1296; idx += 256) {
        int i = idx / 36, j = idx % 36;
        int tm = i >> 4, tn = j >> 4;
        float g  = (tm <= tn) ? G[i * 48 + j] : G[j * 48 + i];
        float mi = SCALEF * csum[i & 3] * invN;
        float mj = SCALEF * csum[j & 3] * invN;
        float cv = (g - (float)NPIX * mi * mj) * norm;
        if (i == j) cv += 1e-5f;
        out[idx] = cv;
    }
}

// ---------------------------------------------------------------------------
// Kernel 2: parallel cyclic Jacobi eigensolver (36x36 SPD) + log reconstruction,
// in-place on the workspace matrix. One workgroup per matrix.
// ---------------------------------------------------------------------------
__device__ __forceinline__ void jacobi_pair(int r, int i, int& p, int& q) {
    // round-robin tournament over 36 players, 35 rounds, 18 disjoint pairs
    if (i == 0) { p = 35; q = r; }
    else        { p = (r + i) % 35; q = (r - i + 70) % 35; }
    if (p > q) { int t = p; p = q; q = t; }
}

__global__ __launch_bounds__(256) void jacobi_log_kernel(float* __restrict__ mats) {
    __shared__ float A[36][37];
    __shared__ float V[36][37];
    __shared__ float cs[18], sn[18];
    __shared__ float llam[36];
    const int tid = threadIdx.x;
    float* __restrict__ M = mats + (size_t)blockIdx.x * 1296;

    for (int idx = tid; idx < 1296; idx += 256) {
        int i = idx / 36, j = idx % 36;
        A[i][j] = M[idx];
        V[i][j] = (i == j) ? 1.0f : 0.0f;
    }
    __syncthreads();

    for (int sweep = 0; sweep < 12; ++sweep) {
        for (int r = 0; r < 35; ++r) {
            if (tid < 18) {
                int p, q; jacobi_pair(r, tid, p, q);
                float apq = A[p][q];
                float c = 1.0f, s = 0.0f;
                if (fabsf(apq) > 1e-30f) {
                    float theta = (A[q][q] - A[p][p]) / (2.0f * apq);
                    float t = copysignf(1.0f, theta) /
                              (fabsf(theta) + sqrtf(1.0f + theta * theta));
                    c = 1.0f / sqrtf(1.0f + t * t);
                    s = t * c;
                }
                cs[tid] = c; sn[tid] = s;
            }
            __syncthreads();
            // rows: A <- J^T A   (disjoint row pairs -> parallel safe)
            for (int tsk = tid; tsk < 18 * 36; tsk += 256) {
                int pi = tsk / 36, j = tsk % 36;
                int p, q; jacobi_pair(r, pi, p, q);
                float c = cs[pi], s = sn[pi];
                float ap = A[p][j], aq = A[q][j];
                A[p][j] = c * ap - s * aq;
                A[q][j] = s * ap + c * aq;
            }
            __syncthreads();
            // cols: A <- A J  and  V <- V J  (disjoint column pairs)
            for (int tsk = tid; tsk < 2 * 18 * 36; tsk += 256) {
                int half = tsk / (18 * 36);
                int rem  = tsk % (18 * 36);
                int pi = rem / 36, i = rem % 36;
                int p, q; jacobi_pair(r, pi, p, q);
                float c = cs[pi], s = sn[pi];
                if (half == 0) {
                    float ap = A[i][p], aq = A[i][q];
                    A[i][p] = c * ap - s * aq;
                    A[i][q] = s * ap + c * aq;
                } else {
                    float ap = V[i][p], aq = V[i][q];
                    V[i][p] = c * ap - s * aq;
                    V[i][q] = s * ap + c * aq;
                }
            }
            __syncthreads();
        }
    }

    if (tid < 36) llam[tid] = __logf(fmaxf(A[tid][tid], 1e-6f));
    __syncthreads();

    // log_cov = V diag(log l) V^T   (reads LDS only -> in-place global write ok)
    for (int idx = tid; idx < 1296; idx += 256) {
        int i = idx / 36, j = idx % 36;
        float sum = 0.0f;
        #pragma unroll 4
        for (int k = 0; k < 36; ++k) sum += V[i][k] * llam[k] * V[j][k];
        M[idx] = sum;
    }
}

// ---------------------------------------------------------------------------
// Kernel 3: mean((logX - logY)^2), single-block tree reduction (deterministic).
// ---------------------------------------------------------------------------
__global__ __launch_bounds__(256) void loss_kernel(const float* __restrict__ logs,
                                                   float* __restrict__ out) {
    __shared__ float red[256];
    const int tid = threadIdx.x;
    const int n = 64 * 1296;
    float partial = 0.0f;
    for (int idx = tid; idx < n; idx += 256) {
        float d = logs[idx] - logs[n + idx];
        partial += d * d;
    }
    red[tid] = partial;
    __syncthreads();
    for (int s = 128; s > 0; s >>= 1) {
        if (tid < s) red[tid] += red[tid + s];
        __syncthreads();
    }
    if (tid == 0) out[0] = red[0] / (float)n;
}

extern "C" void kernel_launch(void* const* d_in, const int* in_sizes, int n_in,
                              void* d_out, int out_size, void* d_ws, size_t ws_size,
                              hipStream_t stream) {
    const float* x = (const float*)d_in[0];
    const float* y = (const float*)d_in[1];
    float* cov = (float*)d_ws;           // 128 matrices * 1296 floats = 663,552 B

    gram_cov_kernel<<<128, 256, 0, stream>>>(x, y, cov);
    jacobi_log_kernel<<<128, 256, 0, stream>>>(cov);
    loss_kernel<<<1, 256, 0, stream>>>(cov, (float*)d_out);
}